// LSTMPred_52106543235777
// MI455X (gfx1250) — compile-verified
//
#include <hip/hip_runtime.h>

// LSTM sizes (compile-time constants from the reference)
#define T_STEPS 512
#define BATCH   8192
#define HID     5
#define GATES   20

typedef __attribute__((ext_vector_type(2))) float v2f;
typedef __attribute__((ext_vector_type(8))) float v8f;

// D(16x16 f32) = A(16x4 f32) * B(4x16 f32) + C ; fp32 CDNA5 WMMA
static __device__ __forceinline__ v8f wmma4(v2f a, v2f b, v8f c) {
  return __builtin_amdgcn_wmma_f32_16x16x4_f32(
      /*neg_a=*/false, a, /*neg_b=*/false, b,
      /*c_mod=*/(short)0, c, /*reuse_a=*/false, /*reuse_b=*/false);
}

// Pure-VALU xor-16 lane swap (v_permlanex16_b32 with identity selects).
static __device__ __forceinline__ float xswap16(float v) {
  int i = __builtin_bit_cast(int, v);
  int r = __builtin_amdgcn_permlanex16(i, i, 0x76543210u, 0xfedcba98u,
                                       /*fi=*/false, /*bound_ctrl=*/false);
  return __builtin_bit_cast(float, r);
}

// Overflow-safe, division-free activations (v_exp_f32 + v_rcp_f32).
static __device__ __forceinline__ float fast_sigmoid(float x) {
  return __builtin_amdgcn_rcpf(1.0f + __expf(-x));
}
static __device__ __forceinline__ float fast_tanh(float x) {
  return 1.0f - 2.0f * __builtin_amdgcn_rcpf(1.0f + __expf(2.0f * x));
}

// A-operand weight chunk: A 16x4 f32 layout -> lane m (0..15 per half) holds
// row m; half h holds K = 4*chunk + 2h, +1.
static __device__ __forceinline__ v2f load_wchunk(const float* __restrict__ W,
                                                  int Kdim, int m, int half, int chunk) {
  int k = chunk * 4 + half * 2;
  v2f r;
  r.x = (m < GATES && (k    ) < Kdim) ? W[m * Kdim + k    ] : 0.0f;
  r.y = (m < GATES && (k + 1) < Kdim) ? W[m * Kdim + k + 1] : 0.0f;
  return r;
}

struct LayerHH {           // hidden projection W_hh (20x5): 2 K-chunks, bias folded
  v2f a[2][2];             // [kchunk][mtile]
  v8f bias[2];             // [mtile], b_ih + b_hh in D layout
};
struct LayerIN3 { v2f a[2]; };      // input projection K=3: 1 chunk, [mtile]
struct LayerIN5 { v2f a[2][2]; };   // input projection K=5: 2 chunks, [kchunk][mtile]

static __device__ __forceinline__ void load_hh(LayerHH& L,
    const float* __restrict__ w_hh,
    const float* __restrict__ b_ih, const float* __restrict__ b_hh,
    int bl, int half) {
#pragma unroll
  for (int mt = 0; mt < 2; ++mt) {
    int m = mt * 16 + bl;
    L.a[0][mt] = load_wchunk(w_hh, HID, m, half, 0);
    L.a[1][mt] = load_wchunk(w_hh, HID, m, half, 1);
    v8f bv;
#pragma unroll
    for (int r = 0; r < 8; ++r) {    // D layout: VGPR r holds M = mt*16 + 8*half + r
      int mb = mt * 16 + half * 8 + r;
      bv[r] = (mb < GATES) ? (b_ih[mb] + b_hh[mb]) : 0.0f;
    }
    L.bias[mt] = bv;
  }
}

// B operand for a 5-vector (K=5 padded to 8): chunk0 = {u0|u2, u1|u3}, chunk1 = {u4|0, 0|0}
struct BOp { v2f b0, b1; };
static __device__ __forceinline__ BOp build_b5(const float u[HID], bool upper) {
  BOp r;
  r.b0.x = upper ? u[2] : u[0];
  r.b0.y = upper ? u[3] : u[1];
  r.b1.x = upper ? 0.0f : u[4];
  r.b1.y = 0.0f;
  return r;
}

// Gather each batch element's 20 gates into every lane and run the LSTM cell
// update (redundant in both halves; keeps EXEC uniform, no cross-lane on B rebuild).
static __device__ __forceinline__ void gate_update(const v8f d[2],
                                                   float h[HID], float cst[HID],
                                                   bool upper) {
  float g[GATES];
#pragma unroll
  for (int r = 0; r < 8; ++r) {
    float own = d[0][r];
    float oth = xswap16(own);
    g[r]     = upper ? oth : own;
    g[8 + r] = upper ? own : oth;
  }
#pragma unroll
  for (int r = 0; r < 4; ++r) {
    float own = d[1][r];             // lower-half rows 0..3 = gates 16..19
    float oth = xswap16(own);
    g[16 + r] = upper ? oth : own;
  }
  // PyTorch gate order: i(0..4) f(5..9) g(10..14) o(15..19)
#pragma unroll
  for (int j = 0; j < HID; ++j) {
    float ig = fast_sigmoid(g[j]);
    float fg = fast_sigmoid(g[5 + j]);
    float gg = fast_tanh(g[10 + j]);
    float og = fast_sigmoid(g[15 + j]);
    cst[j] = fg * cst[j] + ig * gg;
    h[j]   = og * fast_tanh(cst[j]);
  }
}

__global__ __launch_bounds__(256) void lstm3_wmma_kernel(
    const float* __restrict__ x,
    const float* __restrict__ w_ih0, const float* __restrict__ w_hh0,
    const float* __restrict__ b_ih0, const float* __restrict__ b_hh0,
    const float* __restrict__ w_ih1, const float* __restrict__ w_hh1,
    const float* __restrict__ b_ih1, const float* __restrict__ b_hh1,
    const float* __restrict__ w_ih2, const float* __restrict__ w_hh2,
    const float* __restrict__ b_ih2, const float* __restrict__ b_hh2,
    const float* __restrict__ out_w, const float* __restrict__ out_b,
    float* __restrict__ out)
{
  const int lid   = threadIdx.x & 31;
  const int wid   = threadIdx.x >> 5;
  const int wave  = blockIdx.x * (blockDim.x >> 5) + wid;
  const int bl    = lid & 15;          // batch slot within wave tile
  const int half  = lid >> 4;
  const bool upper = (half != 0);
  const int batch = wave * 16 + bl;

  LayerHH LH0, LH1, LH2;
  load_hh(LH0, w_hh0, b_ih0, b_hh0, bl, half);
  load_hh(LH1, w_hh1, b_ih1, b_hh1, bl, half);
  load_hh(LH2, w_hh2, b_ih2, b_hh2, bl, half);

  LayerIN3 LI0;
  LayerIN5 LI1, LI2;
#pragma unroll
  for (int mt = 0; mt < 2; ++mt) {
    int m = mt * 16 + bl;
    LI0.a[mt]    = load_wchunk(w_ih0, 3,   m, half, 0);   // K=3 fits one chunk
    LI1.a[0][mt] = load_wchunk(w_ih1, HID, m, half, 0);
    LI1.a[1][mt] = load_wchunk(w_ih1, HID, m, half, 1);
    LI2.a[0][mt] = load_wchunk(w_ih2, HID, m, half, 0);
    LI2.a[1][mt] = load_wchunk(w_ih2, HID, m, half, 1);
  }

  float h0[HID] = {0,0,0,0,0}, c0[HID] = {0,0,0,0,0};
  float h1[HID] = {0,0,0,0,0}, c1[HID] = {0,0,0,0,0};
  float h2[HID] = {0,0,0,0,0}, c2[HID] = {0,0,0,0,0};

  const float* __restrict__ xb = x + (size_t)batch * T_STEPS * 3;

  for (int t = 0; t < T_STEPS; ++t) {
    if (t + 8 < T_STEPS)
      __builtin_prefetch(&xb[(t + 8) * 3], 0, 0);   // global_prefetch_b8

    float x0v = xb[t * 3 + 0];
    float x1v = xb[t * 3 + 1];
    float x2v = xb[t * 3 + 2];

    // ---- Off-critical-path: all three hidden projections use step t-1 state,
    //      so they issue up front and fill the pipe.
    BOp bh0 = build_b5(h0, upper);
    BOp bh1 = build_b5(h1, upper);
    BOp bh2 = build_b5(h2, upper);
    v8f H0[2], H1[2], H2[2];
#pragma unroll
    for (int mt = 0; mt < 2; ++mt) {
      H0[mt] = wmma4(LH0.a[1][mt], bh0.b1, wmma4(LH0.a[0][mt], bh0.b0, LH0.bias[mt]));
      H1[mt] = wmma4(LH1.a[1][mt], bh1.b1, wmma4(LH1.a[0][mt], bh1.b0, LH1.bias[mt]));
      H2[mt] = wmma4(LH2.a[1][mt], bh2.b1, wmma4(LH2.a[0][mt], bh2.b0, LH2.bias[mt]));
    }

    // ---- Layer 0: one WMMA on the critical path (input K=3).
    v2f bx;
    bx.x = upper ? x2v : x0v;
    bx.y = upper ? 0.0f : x1v;
    v8f d[2];
#pragma unroll
    for (int mt = 0; mt < 2; ++mt) d[mt] = wmma4(LI0.a[mt], bx, H0[mt]);
    gate_update(d, h0, c0, upper);

    // ---- Layer 1: two WMMAs on the critical path (input = h0[t], K=5).
    BOp bu1 = build_b5(h0, upper);
#pragma unroll
    for (int mt = 0; mt < 2; ++mt)
      d[mt] = wmma4(LI1.a[1][mt], bu1.b1, wmma4(LI1.a[0][mt], bu1.b0, H1[mt]));
    gate_update(d, h1, c1, upper);

    // ---- Layer 2: two WMMAs on the critical path (input = h1[t], K=5).
    BOp bu2 = build_b5(h1, upper);
#pragma unroll
    for (int mt = 0; mt < 2; ++mt)
      d[mt] = wmma4(LI2.a[1][mt], bu2.b1, wmma4(LI2.a[0][mt], bu2.b0, H2[mt]));
    gate_update(d, h2, c2, upper);
  }

  float acc = out_b[0];
#pragma unroll
  for (int j = 0; j < HID; ++j) acc += h2[j] * out_w[j];
  if (!upper) out[batch] = acc;        // lanes 0..15 cover all 16 batch slots
}

extern "C" void kernel_launch(void* const* d_in, const int* in_sizes, int n_in,
                              void* d_out, int out_size, void* d_ws, size_t ws_size,
                              hipStream_t stream) {
  (void)in_sizes; (void)n_in; (void)out_size; (void)d_ws; (void)ws_size;
  const float* x     = (const float*)d_in[0];
  const float* w_ih0 = (const float*)d_in[1];
  const float* w_hh0 = (const float*)d_in[2];
  const float* b_ih0 = (const float*)d_in[3];
  const float* b_hh0 = (const float*)d_in[4];
  const float* w_ih1 = (const float*)d_in[5];
  const float* w_hh1 = (const float*)d_in[6];
  const float* b_ih1 = (const float*)d_in[7];
  const float* b_hh1 = (const float*)d_in[8];
  const float* w_ih2 = (const float*)d_in[9];
  const float* w_hh2 = (const float*)d_in[10];
  const float* b_ih2 = (const float*)d_in[11];
  const float* b_hh2 = (const float*)d_in[12];
  const float* out_w = (const float*)d_in[13];
  const float* out_b = (const float*)d_in[14];
  float* out = (float*)d_out;

  // 16 batch elements per wave, 8 waves per block -> 128 batch/block, 64 blocks.
  const int threads = 256;
  const int blocks  = BATCH / (16 * (threads / 32));
  lstm3_wmma_kernel<<<blocks, threads, 0, stream>>>(
      x, w_ih0, w_hh0, b_ih0, b_hh0,
      w_ih1, w_hh1, b_ih1, b_hh1,
      w_ih2, w_hh2, b_ih2, b_hh2,
      out_w, out_b, out);
}